// IM2HT_68977174774305
// MI455X (gfx1250) — compile-verified
//
#include <hip/hip_runtime.h>
#include <stdint.h>

#ifndef __has_builtin
#define __has_builtin(x) 0
#endif

#if __has_builtin(__builtin_amdgcn_global_load_async_to_lds_b32) && \
    __has_builtin(__builtin_amdgcn_s_wait_asynccnt)
#define USE_ASYNC 1
#else
#define USE_ASYNC 0
#endif

#define GAS __attribute__((address_space(1)))
#define LAS __attribute__((address_space(3)))

// gfx12-family CPol: TH in bits[2:0]; TH=1 -> non-temporal (streaming) load.
// Index arrays are use-once (128 MB) -- keep them from evicting the L2-resident
// accumulation volume.
#define CPOL_NT 1

static constexpr int kC       = 32;   // channels == wave32 lanes
static constexpr int kThreads = 256;  // 8 waves / block
static constexpr int kWaves   = kThreads / 32;
static constexpr int kBatch   = 32;   // pairs per wave-batch

// ---------------------------------------------------------------------------
// Transpose feats (C, N_PIX) -> featsT (N_PIX, C) so per-pair channel gathers
// become one coalesced 128B read (keeps non-atomic L2 traffic minimal).
// ---------------------------------------------------------------------------
__global__ void feats_transpose_kernel(const float* __restrict__ feats,
                                       float* __restrict__ featsT, int npix) {
  __shared__ float tile[32][33];
  const int p0 = blockIdx.x * 32;
  const int tx = threadIdx.x;  // 0..31
  const int ty = threadIdx.y;  // 0..7
  for (int c = ty; c < 32; c += 8)
    tile[c][tx] = feats[(size_t)c * npix + (size_t)(p0 + tx)];
  __syncthreads();
  for (int pr = ty; pr < 32; pr += 8)
    featsT[(size_t)(p0 + pr) * 32 + tx] = tile[tx][pr];
}

// ---------------------------------------------------------------------------
// Scatter-accumulate. Lane = channel. Each wave consumes batches of 32 pairs,
// with the index streams double-buffered through LDS via async-to-LDS loads.
// IS/LS: compile-time feature strides (-1 = use runtime values).
// ---------------------------------------------------------------------------
template <long IS, long LS>
__global__ __launch_bounds__(kThreads) void hough_scatter_kernel(
    const float* __restrict__ featsG, long fIdxStrideRt, long fLaneStrideRt,
    const int* __restrict__ voxS, const int* __restrict__ idxS,
    const int* __restrict__ voxD, const int* __restrict__ idxD,
    float* __restrict__ vol, int K, long nVox) {
  const long fIdxStride  = (IS >= 0) ? IS : fIdxStrideRt;
  const long fLaneStride = (LS >= 0) ? LS : fLaneStrideRt;

  const int lane  = threadIdx.x & 31;
  const int wib   = threadIdx.x >> 5;
  const long wave   = (long)blockIdx.x * kWaves + wib;
  const long nwaves = (long)gridDim.x * kWaves;
  const int  nb     = K / kBatch;  // full batches

  float*       volLane  = vol + (long)lane * nVox;
  const float* featLane = featsG + (long)lane * fLaneStride;

#if USE_ASYNC
  // Per wave: 2 buffers x 32 pairs x 4 ints (interleaved so a pair's four
  // indices occupy one 16B slot -> one broadcast ds_load_b128 per pair).
  __shared__ __align__(16) int lds[kWaves * 2 * kBatch * 4];
  int*     mybuf = &lds[wib * (2 * kBatch * 4)];
  LAS int* lbuf  = (LAS int*)mybuf;

  long b   = wave;
  int  cur = 0;
  if (b < nb) {
    const long k0   = b * kBatch + lane;
    LAS int*   slot = lbuf + lane * 4;
    __builtin_amdgcn_global_load_async_to_lds_b32((GAS int*)(voxS + k0), slot + 0, 0, CPOL_NT);
    __builtin_amdgcn_global_load_async_to_lds_b32((GAS int*)(idxS + k0), slot + 1, 0, CPOL_NT);
    __builtin_amdgcn_global_load_async_to_lds_b32((GAS int*)(voxD + k0), slot + 2, 0, CPOL_NT);
    __builtin_amdgcn_global_load_async_to_lds_b32((GAS int*)(idxD + k0), slot + 3, 0, CPOL_NT);
  }
  for (; b < nb; b += nwaves) {
    const long bn = b + nwaves;
    if (bn < nb) {
      const long k0   = bn * kBatch + lane;
      LAS int*   slot = lbuf + (cur ^ 1) * (kBatch * 4) + lane * 4;
      __builtin_amdgcn_global_load_async_to_lds_b32((GAS int*)(voxS + k0), slot + 0, 0, CPOL_NT);
      __builtin_amdgcn_global_load_async_to_lds_b32((GAS int*)(idxS + k0), slot + 1, 0, CPOL_NT);
      __builtin_amdgcn_global_load_async_to_lds_b32((GAS int*)(voxD + k0), slot + 2, 0, CPOL_NT);
      __builtin_amdgcn_global_load_async_to_lds_b32((GAS int*)(idxD + k0), slot + 3, 0, CPOL_NT);
      __builtin_amdgcn_s_wait_asynccnt(4);  // current buffer's 4 loads done
    } else {
      __builtin_amdgcn_s_wait_asynccnt(0);
    }
    __asm__ __volatile__("" ::: "memory");
    const int4* q = (const int4*)(mybuf + cur * (kBatch * 4));
#pragma unroll 4
    for (int j = 0; j < kBatch; ++j) {
      const int4  p  = q[j];  // {vox_s, idx_s, vox_d, idx_d} broadcast to wave
      const float fs = featLane[(long)p.y * fIdxStride];
      const float fd = featLane[(long)p.w * fIdxStride];
      atomicAdd(&volLane[p.x], fs);
      atomicAdd(&volLane[p.z], fd);
    }
    cur ^= 1;
  }
#else
  // Fallback: coalesced register loads + shuffle broadcast.
  for (long b = wave; b < nb; b += nwaves) {
    const long k0 = b * kBatch + lane;
    const int mvs = voxS[k0], mis = idxS[k0], mvd = voxD[k0], mid = idxD[k0];
    for (int j = 0; j < kBatch; ++j) {
      const int vs = __shfl(mvs, j, 32);
      const int is = __shfl(mis, j, 32);
      const int vd = __shfl(mvd, j, 32);
      const int id = __shfl(mid, j, 32);
      atomicAdd(&volLane[vs], featLane[(long)is * fIdxStride]);
      atomicAdd(&volLane[vd], featLane[(long)id * fIdxStride]);
    }
  }
#endif

  // Remainder pairs (K % 32) — handled by a single wave (K=8M is a multiple
  // of 32, so this is normally empty; kept for generality).
  if (wave == 0) {
    for (long k = (long)nb * kBatch; k < (long)K; ++k) {
      const int vs = voxS[k], is = idxS[k], vd = voxD[k], id = idxD[k];
      atomicAdd(&volLane[vs], featLane[(long)is * fIdxStride]);
      atomicAdd(&volLane[vd], featLane[(long)id * fIdxStride]);
    }
  }
}

// ---------------------------------------------------------------------------
// Inputs (setup_inputs order): feats_src_dst(C*N_PIX f32), voxels_src(K i32),
// voxels_dst(K i32), idxs_src(K i32), idxs_dst(K i32), h, w, d (scalars).
// Output: vol (C, H, W, D) f32 == (C, n_vox).
// ---------------------------------------------------------------------------
extern "C" void kernel_launch(void* const* d_in, const int* in_sizes, int n_in,
                              void* d_out, int out_size, void* d_ws, size_t ws_size,
                              hipStream_t stream) {
  (void)n_in;
  const float* feats = (const float*)d_in[0];
  const int*   voxS  = (const int*)d_in[1];
  const int*   voxD  = (const int*)d_in[2];
  const int*   idxS  = (const int*)d_in[3];
  const int*   idxD  = (const int*)d_in[4];

  const int  K    = in_sizes[1];
  const int  npix = in_sizes[0] / kC;
  const long nVox = (long)out_size / kC;

  (void)hipMemsetAsync(d_out, 0, (size_t)out_size * sizeof(float), stream);

  const size_t tBytes = (size_t)npix * kC * sizeof(float);
  const int    blocks = 2048;  // 16384 waves -> ~15 batches each, deep pipeline

  if (ws_size >= tBytes && (npix % 32) == 0) {
    // Transposed-feature fast path: featsT[p*32 + c], compile-time strides.
    float* featsT = (float*)d_ws;
    dim3 tb(32, 8);
    feats_transpose_kernel<<<npix / 32, tb, 0, stream>>>(feats, featsT, npix);
    hough_scatter_kernel<kC, 1><<<blocks, kThreads, 0, stream>>>(
        featsT, kC, 1, voxS, idxS, voxD, idxD, (float*)d_out, K, nVox);
  } else {
    // Generic path: feats[c*npix + p], runtime strides.
    hough_scatter_kernel<-1, -1><<<blocks, kThreads, 0, stream>>>(
        feats, 1, npix, voxS, idxS, voxD, idxD, (float*)d_out, K, nVox);
  }
}